// deeds_69260642615549
// MI455X (gfx1250) — compile-verified
//
#include <hip/hip_runtime.h>
#include <math.h>

// ---------------- constants ----------------
#define CCH   16
#define DIMV  64
#define GSZ   32
#define DWN   15
#define D3    3375            // 15^3
#define G3    32768           // 32^3
#define NB    26              // neighborhood extent per axis (offsets -12..13)
#define NB3   (NB*NB*NB)      // 17576
#define CHVOL (DIMV*DIMV*DIMV) // 262144
#define KPAD  3376            // D3 padded to a multiple of 4 for the WMMA K loop

typedef float v2f __attribute__((ext_vector_type(2)));
typedef float v8f __attribute__((ext_vector_type(8)));

__device__ __forceinline__ int clampi(int v, int lo, int hi) {
    return v < lo ? lo : (v > hi ? hi : v);
}

// =====================================================================
// Kernel 1: deeds_cost(p, d) = alpha1 + alpha0 * SSD over 16 channels
// One workgroup per grid point p. 26^3 neighborhood of feat50 staged in
// LDS per channel; each thread owns ~14 displacement accumulators whose
// trilinear weights / LDS base offsets are hoisted out of the channel
// loop (they depend only on d). Next-channel neighborhood is prefetched.
// =====================================================================
__global__ __launch_bounds__(256) void k_deeds(const float* __restrict__ f0,
                                               const float* __restrict__ f5,
                                               const float* __restrict__ alpha,
                                               float* __restrict__ out)
{
    extern __shared__ float smem[];
    float* neigh  = smem;                 // 17576
    float* fixedc = smem + NB3;           // 16
    float* wtab   = smem + NB3 + 16;      // 45  (frac weights, axis-major: x,y,z)
    int*   otab   = (int*)(smem + NB3 + 16 + 45); // 45 (relative offsets)

    const int p  = blockIdx.x;
    const int iz = p >> 10, iy = (p >> 5) & 31, ix = p & 31;
    const int tid = threadIdx.x;

    // per-axis offset/weight tables: replicate reference fp32 coordinate math
    if (tid < 45) {
        int axis = tid / 15, j = tid - axis * 15;
        int ib = (axis == 0) ? ix : (axis == 1 ? iy : iz);
        float g = (2.0f * (float)ib + 1.0f) / 32.0f - 1.0f;
        float s = 0.4f * ((2.0f * (float)j + 1.0f) / 15.0f - 1.0f);
        float fi = (((g + s) + 1.0f) * 64.0f - 1.0f) * 0.5f;
        float fl = floorf(fi);
        wtab[tid] = fi - fl;
        otab[tid] = (int)fl - 2 * ib;     // in [-12, 12]
    }
    // fixed feature: trilinear at 2i+0.5 exactly == average of 2x2x2 block
    if (tid < CCH) {
        const float* b = f0 + tid * CHVOL + (2 * iz) * 4096 + (2 * iy) * 64 + (2 * ix);
        float s = b[0] + b[1] + b[64] + b[65] + b[4096] + b[4097] + b[4160] + b[4161];
        fixedc[tid] = 0.125f * s;
    }
    __syncthreads();

    // hoist per-displacement decode out of the channel loop
    float wxv[14], wyv[14], wzv[14];
    int   basev[14];
    float acc[14];
#pragma unroll
    for (int k = 0; k < 14; ++k) {
        acc[k] = 0.0f;
        wxv[k] = 0.0f; wyv[k] = 0.0f; wzv[k] = 0.0f; basev[k] = 0;
        int d = tid + (k << 8);
        if (d < D3) {
            int jz = d / 225;
            int r  = d - jz * 225;
            int jy = r / 15;
            int jx = r - jy * 15;
            wxv[k] = wtab[jx];
            wyv[k] = wtab[15 + jy];
            wzv[k] = wtab[30 + jz];
            basev[k] = (otab[30 + jz] + 12) * (NB * NB)
                     + (otab[15 + jy] + 12) * NB
                     + (otab[jx] + 12);
        }
    }

    for (int c = 0; c < CCH; ++c) {
        __syncthreads();
        const float* fc = f5 + c * CHVOL;
        for (int idx = tid; idx < NB3; idx += 256) {
            int uz = idx / (NB * NB);
            int r  = idx - uz * (NB * NB);
            int uy = r / NB;
            int ux = r - uy * NB;
            int az = clampi(2 * iz + uz - 12, 0, 63);
            int ay = clampi(2 * iy + uy - 12, 0, 63);
            int ax = clampi(2 * ix + ux - 12, 0, 63);
            int addr = az * 4096 + ay * 64 + ax;
            neigh[idx] = fc[addr];
            if (c + 1 < CCH) __builtin_prefetch(fc + CHVOL + addr, 0, 0);
        }
        __syncthreads();
        float fv = fixedc[c];
#pragma unroll
        for (int k = 0; k < 14; ++k) {
            int d = tid + (k << 8);
            if (d < D3) {
                int base = basev[k];
                float wx = wxv[k], wy = wyv[k], wz = wzv[k];
                float n000 = neigh[base],                n001 = neigh[base + 1];
                float n010 = neigh[base + NB],           n011 = neigh[base + NB + 1];
                float n100 = neigh[base + NB * NB],      n101 = neigh[base + NB * NB + 1];
                float n110 = neigh[base + NB * NB + NB], n111 = neigh[base + NB * NB + NB + 1];
                float a00 = n000 + wx * (n001 - n000);
                float a01 = n010 + wx * (n011 - n010);
                float a10 = n100 + wx * (n101 - n100);
                float a11 = n110 + wx * (n111 - n110);
                float lo = a00 + wy * (a01 - a00);
                float hi = a10 + wy * (a11 - a10);
                float m  = lo + wz * (hi - lo);
                float df = fv - m;
                acc[k] = fmaf(df, df, acc[k]);
            }
        }
    }

    float a0 = alpha[0], a1 = alpha[1];
    long long rowoff = (long long)p * D3;
#pragma unroll
    for (int k = 0; k < 14; ++k) {
        int d = tid + (k << 8);
        if (d < D3) out[rowoff + d] = a1 + a0 * acc[k];
    }
}

// =====================================================================
// Kernel 2: regularize over the 15^3 displacement cube per grid point:
// edge-pad(3) -> min-pool 3^3 -> avg 3^3 -> avg 3^3 (all clamped-index)
// =====================================================================
__global__ __launch_bounds__(256) void k_regularize(const float* __restrict__ in,
                                                    float* __restrict__ out)
{
    extern __shared__ float smem[];
    float* x  = smem;                  // 15^3 = 3375
    float* m  = smem + 3375;           // 19^3 = 6859
    float* a1 = smem + 3375 + 6859;    // 17^3 = 4913
    const int p = blockIdx.x, tid = threadIdx.x;
    const float* row = in + (long long)p * D3;

    for (int i = tid; i < D3; i += 256) x[i] = row[i];
    __syncthreads();

    // min-pool: M[v] = min over X[cl(v-3..v-1)] per axis
    for (int i = tid; i < 6859; i += 256) {
        int z = i / 361; int r = i - z * 361; int y = r / 19; int xx = r - y * 19;
        float mn = 3.402823466e38f;
        for (int tz = -3; tz <= -1; ++tz) {
            int zz = clampi(z + tz, 0, 14);
            for (int ty = -3; ty <= -1; ++ty) {
                int yy = clampi(y + ty, 0, 14);
                int b = (zz * 15 + yy) * 15;
                for (int tx = -3; tx <= -1; ++tx) {
                    int xc = clampi(xx + tx, 0, 14);
                    mn = fminf(mn, x[b + xc]);
                }
            }
        }
        m[i] = mn;
    }
    __syncthreads();

    for (int i = tid; i < 4913; i += 256) {
        int z = i / 289; int r = i - z * 289; int y = r / 17; int xx = r - y * 17;
        float s = 0.0f;
        for (int tz = 0; tz < 3; ++tz)
            for (int ty = 0; ty < 3; ++ty)
                for (int tx = 0; tx < 3; ++tx)
                    s += m[((z + tz) * 19 + (y + ty)) * 19 + (xx + tx)];
        a1[i] = s * (1.0f / 27.0f);
    }
    __syncthreads();

    float* orow = out + (long long)p * D3;
    for (int i = tid; i < D3; i += 256) {
        int z = i / 225; int r = i - z * 225; int y = r / 15; int xx = r - y * 15;
        float s = 0.0f;
        for (int tz = 0; tz < 3; ++tz)
            for (int ty = 0; ty < 3; ++ty)
                for (int tx = 0; tx < 3; ++tx)
                    s += a1[((z + tz) * 17 + (y + ty)) * 17 + (xx + tx)];
        orow[i] = s * (1.0f / 27.0f);
    }
}

// =====================================================================
// Kernel 3: one separable axis pass of diffuse: two clamped avg-3s over
// the 32^3 grid == clamped 5-tap [1,2,3,2,1]/9 per axis, applied x,y,z.
// =====================================================================
__global__ __launch_bounds__(256) void k_diffuse_pass(const float* __restrict__ in,
                                                      float* __restrict__ out,
                                                      int axis, long long ntot)
{
    long long e = (long long)blockIdx.x * 256 + threadIdx.x;
    if (e >= ntot) return;
    long long pl = e / D3;
    int p = (int)pl;
    int iz = p >> 10, iy = (p >> 5) & 31, ix = p & 31;
    int v = (axis == 0) ? ix : (axis == 1 ? iy : iz);
    long long stride = ((axis == 0) ? 1LL : (axis == 1 ? 32LL : 1024LL)) * D3;
    long long base = e - (long long)v * stride;
    float s = 0.0f;
    const float K[5] = {1.0f, 2.0f, 3.0f, 2.0f, 1.0f};
#pragma unroll
    for (int a = 0; a < 5; ++a) {
        int u = clampi(v + a - 2, 0, 31);
        s += K[a] * in[base + (long long)u * stride];
    }
    out[e] = s * (1.0f / 9.0f);
}

// =====================================================================
// Kernel 4: cost2 = alpha4 + alpha2 * deeds_cost + alpha3 * cost_avg
// =====================================================================
__global__ __launch_bounds__(256) void k_combine(const float* __restrict__ dc,
                                                 const float* __restrict__ ca,
                                                 const float* __restrict__ alpha,
                                                 float* __restrict__ out, long long ntot)
{
    long long e = (long long)blockIdx.x * 256 + threadIdx.x;
    if (e >= ntot) return;
    out[e] = alpha[4] + alpha[2] * dc[e] + alpha[3] * ca[e];
}

// =====================================================================
// Kernel 5: row softmax of (-alpha5 * cost_avg) over 3375 displacements
// =====================================================================
__global__ __launch_bounds__(256) void k_softmax(const float* __restrict__ in,
                                                 const float* __restrict__ alpha,
                                                 float* __restrict__ out)
{
    __shared__ float red[256];
    const int p = blockIdx.x, tid = threadIdx.x;
    const float a5 = alpha[5];
    const float* row = in + (long long)p * D3;

    float v[14];
    float mx = -3.402823466e38f;
#pragma unroll
    for (int k = 0; k < 14; ++k) {
        int d = tid + (k << 8);
        v[k] = (d < D3) ? (-a5 * row[d]) : -3.402823466e38f;
        mx = fmaxf(mx, v[k]);
    }
    red[tid] = mx; __syncthreads();
    for (int s = 128; s > 0; s >>= 1) {
        if (tid < s) red[tid] = fmaxf(red[tid], red[tid + s]);
        __syncthreads();
    }
    mx = red[0]; __syncthreads();

    float sm = 0.0f;
#pragma unroll
    for (int k = 0; k < 14; ++k) {
        int d = tid + (k << 8);
        if (d < D3) { v[k] = expf(v[k] - mx); sm += v[k]; }
    }
    red[tid] = sm; __syncthreads();
    for (int s = 128; s > 0; s >>= 1) {
        if (tid < s) red[tid] += red[tid + s];
        __syncthreads();
    }
    float inv = 1.0f / red[0];

    float* orow = out + (long long)p * D3;
#pragma unroll
    for (int k = 0; k < 14; ++k) {
        int d = tid + (k << 8);
        if (d < D3) orow[d] = v[k] * inv;
    }
}

// =====================================================================
// Kernel 6: pred = cost_soft (32768 x 3375) @ SHIFT (3375 x 3), fp32,
// via V_WMMA_F32_16X16X4_F32. One wave per 16-row tile. SHIFT operand
// staged once in LDS as a 4x3376 table (rows 0-2 = x/y/z components,
// row 3 = zeros for padded N columns; entry k=3375 = 0 for the K tail),
// so the K loop is branch-free: 2 x b32 A-loads, 1 x b64 LDS B-load,
// 1 x v_wmma per chunk. The one-element A overread per row multiplies
// the guaranteed-zero B row, contributing exactly 0.
// =====================================================================
__device__ __forceinline__ float shift_comp(int dd, int n)
{
    if (dd >= D3 || n >= 3) return 0.0f;
    int j;
    if (n == 0)      j = dd % 15;
    else if (n == 1) j = (dd / 15) % 15;
    else             j = dd / 225;
    return 0.4f * ((2.0f * (float)j + 1.0f) / 15.0f - 1.0f);
}

__global__ __launch_bounds__(32) void k_pred_wmma(const float* __restrict__ cs,
                                                  float* __restrict__ pred)
{
    __shared__ float sh[4 * KPAD];    // 54,016 bytes
    const int lane = threadIdx.x;
    const int r0   = blockIdx.x * 16;
    const int mrow = lane & 15;
    const int half = lane >> 4;       // 0: K pair {0,1}; 1: K pair {2,3}

    // build padded SHIFT table once (all div/mod cost lives here)
    for (int i = lane; i < 4 * KPAD; i += 32) {
        int n = i / KPAD;
        int k = i - n * KPAD;
        sh[i] = shift_comp(k, n);
    }
    __syncthreads();

    const float* rowp = cs + (long long)(r0 + mrow) * D3 + 2 * half;
    const float* brow = sh + (mrow < 3 ? mrow : 3) * KPAD + 2 * half;

    v8f c;
#pragma unroll
    for (int i = 0; i < 8; ++i) c[i] = 0.0f;

    for (int kk = 0; kk < KPAD; kk += 4) {
        v2f a, b;
        a.x = rowp[kk];
        a.y = rowp[kk + 1];
        b = *(const v2f*)(brow + kk);     // 8-byte aligned LDS pair
        c = __builtin_amdgcn_wmma_f32_16x16x4_f32(false, a, false, b,
                                                  (short)0, c, false, false);
    }

    if (mrow < 3) {
#pragma unroll
        for (int r = 0; r < 8; ++r) {
            int row = r0 + r + half * 8;  // C/D: vgpr r -> M=r / r+8
            pred[row * 3 + mrow] = c[r];
        }
    }
}

// =====================================================================
// Launch: A,B in d_ws (2 x 442.4 MB); cost_soft region of d_out doubles
// as the third ping-pong buffer until the softmax finally writes it.
// =====================================================================
extern "C" void kernel_launch(void* const* d_in, const int* in_sizes, int n_in,
                              void* d_out, int out_size, void* d_ws, size_t ws_size,
                              hipStream_t stream)
{
    const float* f0    = (const float*)d_in[0];
    const float* f5    = (const float*)d_in[1];
    const float* alpha = (const float*)d_in[2];
    float* out = (float*)d_out;

    const long long Ntot = (long long)G3 * D3;      // 110,592,000
    float* A  = (float*)d_ws;
    float* B  = A + Ntot;
    float* Cs = out;                                // cost_soft region as scratch

    const size_t smem1 = (NB3 + 16 + 45 + 45) * sizeof(float);      // ~70.7 KB
    const size_t smem2 = (3375 + 6859 + 4913) * sizeof(float);      // ~60.6 KB
    const int nbE = (int)((Ntot + 255) / 256);                      // 432000

    // 1) deeds_cost -> A
    k_deeds<<<G3, 256, smem1, stream>>>(f0, f5, alpha, A);
    // 2) regularize(A) -> B
    k_regularize<<<G3, 256, smem2, stream>>>(A, B);
    // 3) diffuse(B): separable x,y,z -> cost_avg1 in Cs
    k_diffuse_pass<<<nbE, 256, 0, stream>>>(B, Cs, 0, Ntot);
    k_diffuse_pass<<<nbE, 256, 0, stream>>>(Cs, B, 1, Ntot);
    k_diffuse_pass<<<nbE, 256, 0, stream>>>(B, Cs, 2, Ntot);
    // 4) cost2 = a4 + a2*A + a3*Cs -> B
    k_combine<<<nbE, 256, 0, stream>>>(A, Cs, alpha, B, Ntot);
    // 5) regularize(B) -> A
    k_regularize<<<G3, 256, smem2, stream>>>(B, A);
    // 6) diffuse(A) -> cost_avg2 in A
    k_diffuse_pass<<<nbE, 256, 0, stream>>>(A, B, 0, Ntot);
    k_diffuse_pass<<<nbE, 256, 0, stream>>>(B, Cs, 1, Ntot);
    k_diffuse_pass<<<nbE, 256, 0, stream>>>(Cs, A, 2, Ntot);
    // 7) softmax rows -> cost_soft (d_out[0..Ntot))
    k_softmax<<<G3, 256, 0, stream>>>(A, alpha, out);
    // 8) pred = cost_soft @ SHIFT via WMMA fp32 -> d_out[Ntot..)
    k_pred_wmma<<<G3 / 16, 32, 0, stream>>>(out, out + Ntot);
}